// GCNModel2_89773406421367
// MI455X (gfx1250) — compile-verified
//
#include <hip/hip_runtime.h>
#include <hip/hip_bf16.h>
#include <math.h>

// ---------------------------------------------------------------------------
// GCN on MI455X (gfx1250, wave32). Whole working set (~130MB) fits in the
// 192MB L2 -> edge scatter/atomic bound. GEMMs use full-precision f32 WMMA
// (V_WMMA_F32_16X16X4_F32) with B pre-packed into fragment layout and an
// explicitly software-pipelined inner loop (loads for k+4 in flight while
// WMMAs consume k) so waits are partial, not wait-to-zero.
// ---------------------------------------------------------------------------

#define NN      50000          // nodes (exactly 3125 * 16 = 625 * 80)
#define STRIDE  256            // padded feature stride (fits 128 and 250)
#define MT      5              // M-tiles (16 rows each) per wave strip

typedef float v2f __attribute__((ext_vector_type(2)));
typedef float v8f __attribute__((ext_vector_type(8)));

// ------------------------------- helpers -----------------------------------

__global__ void zero_f32(float* __restrict__ p, int n) {
    int i = blockIdx.x * blockDim.x + threadIdx.x;
    if (i < n) p[i] = 0.0f;
}

__global__ void deg_count(const long long* __restrict__ dst, float* deg, int E) {
    int i = blockIdx.x * blockDim.x + threadIdx.x;
    if (i < E) atomicAdd(&deg[(int)dst[i]], 1.0f);
}

// in-place: deg -> 1/sqrt(deg + 1)  (self loop)
__global__ void finish_dinv(float* __restrict__ deg, int n) {
    int i = blockIdx.x * blockDim.x + threadIdx.x;
    if (i < n) deg[i] = rsqrtf(deg[i] + 1.0f);
}

__global__ void relu_ip(float* __restrict__ p, int n) {
    int i = blockIdx.x * blockDim.x + threadIdx.x;
    if (i < n) p[i] = fmaxf(p[i], 0.0f);
}

// out[row, col] = H[row,col]*dinv[row]^2 + bias[col]   (col <  width)
//              = 0                                     (col >= width, keeps pad 0)
__global__ void init_agg(const float* __restrict__ H,
                         const float* __restrict__ dinv,
                         const float* __restrict__ bias,
                         float* __restrict__ out, int width) {
    int i = blockIdx.x * blockDim.x + threadIdx.x;
    if (i >= NN * STRIDE) return;
    int row = i >> 8;          // / STRIDE
    int col = i & (STRIDE - 1);
    float v = 0.0f;
    if (col < width) {
        float di = dinv[row];
        v = H[i] * di * di + bias[col];
    }
    out[i] = v;
}

// One edge per wave32: coalesced row gather + 32-wide f32 atomic stream to L2.
__global__ void edge_scatter(const long long* __restrict__ srcs,
                             const long long* __restrict__ dsts,
                             const float* __restrict__ H,
                             const float* __restrict__ dinv,
                             float* __restrict__ agg, int E, int width) {
    int wid = blockIdx.x * (blockDim.x >> 5) + (threadIdx.x >> 5);
    if (wid >= E) return;
    int lane = threadIdx.x & 31;
    int s = (int)srcs[wid];
    int d = (int)dsts[wid];
    float norm = dinv[s] * dinv[d];
    const float* hrow = H   + (size_t)s * STRIDE;
    float*       arow = agg + (size_t)d * STRIDE;
    if (width == 128) {
        float4 v = ((const float4*)hrow)[lane];     // 512B/wave coalesced
        float* a = arow + (lane << 2);
        atomicAdd(a + 0, v.x * norm);
        atomicAdd(a + 1, v.y * norm);
        atomicAdd(a + 2, v.z * norm);
        atomicAdd(a + 3, v.w * norm);
    } else {
        for (int c = lane; c < width; c += 32)
            atomicAdd(arow + c, hrow[c] * norm);
    }
}

// ------------------------- B fragment pre-pack ------------------------------
// WMMA f32 16x16x4 B-frag: lanes 0-15 carry K rows {k,k+1}, lanes 16-31 carry
// {k+2,k+3}, N striped across the half-wave. Pack so each lane reads ONE
// contiguous float2:  Bp2[( (k>>2)*2 + hi ) * Npad + n] = {B(k0,n), B(k0+1,n)}
// with k0 = k + 2*hi. Zero-pads K>=Kvalid / N>=Nvalid; optional transpose.
__global__ void pack_b(const float* __restrict__ B, int ldb, int transB,
                       int Kvalid, int Nvalid, int Kpad, int Npad,
                       float2* __restrict__ Bp2) {
    int idx = blockIdx.x * blockDim.x + threadIdx.x;
    int total = (Kpad >> 1) * Npad;              // rows r = (k>>2)*2+hi
    if (idx >= total) return;
    int n  = idx % Npad;
    int r  = idx / Npad;
    int k0 = ((r >> 1) << 2) + ((r & 1) << 1);
    float b0 = 0.0f, b1 = 0.0f;
    if (n < Nvalid) {
        if (!transB) {
            if (k0     < Kvalid) b0 = B[(size_t)k0       * ldb + n];
            if (k0 + 1 < Kvalid) b1 = B[(size_t)(k0 + 1) * ldb + n];
        } else {                                  // B = W^T : B(k,n) = W(n,k)
            if (k0     < Kvalid) b0 = B[(size_t)n * ldb + k0];
            if (k0 + 1 < Kvalid) b1 = B[(size_t)n * ldb + k0 + 1];
        }
    }
    float2 v; v.x = b0; v.y = b1;
    Bp2[idx] = v;
}

// ------------------------------- WMMA GEMM ---------------------------------
// C[M x Npad] = A[M x K] * Bpacked (+ bias). One wave computes a (MT*16) x 16
// strip: one B fragment feeds MT WMMAs. M = 50000 = 625 strips of 80 rows.
// Software-pipelined: iteration k issues the 6 global_load_b64 for k+4 into
// fresh registers, THEN runs the MT WMMAs on the k fragments -> the wait in
// front of the WMMA batch only drains the previous loads (partial loadcnt
// wait) while the next set stays in flight.
__global__ void gemm_wmma_f32(const float* __restrict__ A, int lda,
                              const float* __restrict__ Bp,
                              int Npad, const float* __restrict__ bias,
                              float* __restrict__ C, int ldc,
                              int K, int Nvalid, int tilesN, int totalWaves) {
    int wid = blockIdx.x * (blockDim.x >> 5) + (threadIdx.x >> 5);
    if (wid >= totalWaves) return;              // uniform per wave
    int lane  = threadIdx.x & 31;
    int lh    = lane & 15;
    int hi    = lane >> 4;                      // 0 = lanes 0-15, 1 = 16-31
    int strip = wid / tilesN;
    int tileN = wid - strip * tilesN;
    int m0   = strip * (MT * 16);
    int bcol = (tileN << 4) + lh;

    // loop-invariant bases; per-iteration offsets are small immediates
    const float* Abase = A + (size_t)(m0 + lh) * lda + (hi << 1);
    const size_t rstep = (size_t)16 * lda;      // 16 rows between sub-tiles
    const v2f*   Bbase = (const v2f*)Bp + (size_t)hi * Npad + bcol;
    const size_t bstep = 2 * (size_t)Npad;      // per k-step in float2 units

    v8f acc[MT];
#pragma unroll
    for (int t = 0; t < MT; ++t) { v8f z = {}; acc[t] = z; }

    // ---- pipeline prologue: fragments for k = 0 ----
    v2f bcur = Bbase[0];
    v2f acur[MT];
#pragma unroll
    for (int t = 0; t < MT; ++t)
        acur[t] = *(const v2f*)(Abase + (size_t)t * rstep);

    // ---- steady state: prefetch k+4, compute k ----
#pragma unroll 2
    for (int k = 0; k + 4 < K; k += 4) {
        v2f bnext = Bbase[(size_t)((k >> 2) + 1) * bstep];
        v2f anext[MT];
#pragma unroll
        for (int t = 0; t < MT; ++t)
            anext[t] = *(const v2f*)(Abase + (size_t)t * rstep + (k + 4));
#pragma unroll
        for (int t = 0; t < MT; ++t)
            acc[t] = __builtin_amdgcn_wmma_f32_16x16x4_f32(
                /*neg_a=*/false, acur[t], /*neg_b=*/false, bcur,
                /*c_mod=*/(short)0, acc[t], /*reuse_a=*/false, /*reuse_b=*/false);
        bcur = bnext;
#pragma unroll
        for (int t = 0; t < MT; ++t) acur[t] = anext[t];
    }

    // ---- epilogue: last k-step ----
#pragma unroll
    for (int t = 0; t < MT; ++t)
        acc[t] = __builtin_amdgcn_wmma_f32_16x16x4_f32(
            /*neg_a=*/false, acur[t], /*neg_b=*/false, bcur,
            /*c_mod=*/(short)0, acc[t], /*reuse_a=*/false, /*reuse_b=*/false);

    if (bcol < Nvalid) {                         // only clips for ldc=250 store
        float bv = bias ? bias[bcol] : 0.0f;
#pragma unroll
        for (int t = 0; t < MT; ++t) {
            int mbase = m0 + t * 16 + (hi << 3);
#pragma unroll
            for (int g = 0; g < 8; ++g)
                C[(size_t)(mbase + g) * ldc + bcol] = acc[t][g] + bv;
        }
    }
}

// ------------------------------- launch ------------------------------------

extern "C" void kernel_launch(void* const* d_in, const int* in_sizes, int n_in,
                              void* d_out, int out_size, void* d_ws, size_t ws_size,
                              hipStream_t stream) {
    const float* x    = (const float*)d_in[0];
    const float* W1   = (const float*)d_in[1];
    const float* b1   = (const float*)d_in[2];
    const float* W2   = (const float*)d_in[3];
    const float* b2   = (const float*)d_in[4];
    const float* W3   = (const float*)d_in[5];
    const float* b3   = (const float*)d_in[6];
    const float* W4   = (const float*)d_in[7];
    const float* b4   = (const float*)d_in[8];
    const float* W5   = (const float*)d_in[9];
    const float* b5   = (const float*)d_in[10];
    const float* linW = (const float*)d_in[11];
    const float* linB = (const float*)d_in[12];
    const long long* ei = (const long long*)d_in[13];   // [2, E] int64
    int E = in_sizes[13] / 2;
    const long long* esrc = ei;
    const long long* edst = ei + E;

    // workspace: dinv | Bpack(256KB) | Htmp[NN*256] | featA[NN*256]  (~103 MB)
    float*  dinv  = (float*)d_ws;
    float2* Bpack = (float2*)((char*)d_ws + 200704);
    float*  Htmp  = (float*)((char*)d_ws + 462848);
    float*  featA = Htmp + (size_t)NN * STRIDE;

    // degree -> dinv = 1/sqrt(deg_dst + 1)
    zero_f32   <<<(NN + 255) / 256, 256, 0, stream>>>(dinv, NN);
    deg_count  <<<(E  + 255) / 256, 256, 0, stream>>>(edst, dinv, E);
    finish_dinv<<<(NN + 255) / 256, 256, 0, stream>>>(dinv, NN);

    auto run_gemm = [&](const float* Ain, int lda, const float* W, int ldb,
                        int transB, int Kpad, int Kvalid, int Npad, int Nvalid,
                        const float* bias, float* C, int ldc) {
        int packN = (Kpad >> 1) * Npad;
        pack_b<<<(packN + 255) / 256, 256, 0, stream>>>(
            W, ldb, transB, Kvalid, Nvalid, Kpad, Npad, Bpack);
        int tilesN = Npad >> 4;
        int total  = (NN / (MT * 16)) * tilesN;          // 625 strips
        int blocks = (total + 7) / 8;                    // 8 waves / block
        gemm_wmma_f32<<<blocks, 256, 0, stream>>>(
            Ain, lda, (const float*)Bpack, Npad, bias, C, ldc,
            Kpad, Nvalid, tilesN, total);
    };

    auto run_layer = [&](const float* Ain, int lda, const float* W, int ldb,
                         const float* bias, int K, int Dout, int DoutPad,
                         bool relu) {
        // H = A @ W  (Nvalid = DoutPad, so pad cols of Htmp are not zeroed)
        run_gemm(Ain, lda, W, ldb, /*transB=*/0, K, K, DoutPad, DoutPad,
                 nullptr, Htmp, STRIDE);
        init_agg<<<(NN * STRIDE + 255) / 256, 256, 0, stream>>>(
            Htmp, dinv, bias, featA, Dout);
        edge_scatter<<<(E + 7) / 8, 256, 0, stream>>>(
            esrc, edst, Htmp, dinv, featA, E, Dout);
        if (relu)
            relu_ip<<<(NN * STRIDE + 255) / 256, 256, 0, stream>>>(
                featA, NN * STRIDE);
    };

    run_layer(x,     128,    W1, 128, b1, 128, 128, 128, true);
    run_layer(featA, STRIDE, W2, 128, b2, 128, 128, 128, false);
    run_layer(featA, STRIDE, W3, 128, b3, 128, 128, 128, false);
    run_layer(featA, STRIDE, W4, 128, b4, 128, 128, 128, false);
    run_layer(featA, STRIDE, W5, 250, b5, 128, 250, 256, false);

    // final linear: out = featA[:, :250] @ linW^T + linB
    // featA pad cols (250..255) are zero, so run K=256 unguarded; store clips
    // to ldc = 250 into d_out.
    run_gemm(featA, STRIDE, linW, 250, /*transB=*/1,
             /*Kpad=*/256, /*Kvalid=*/250, /*Npad=*/256, /*Nvalid=*/250,
             linB, (float*)d_out, 250);
}